// Euclidean_83657372991990
// MI455X (gfx1250) — compile-verified
//
#include <hip/hip_runtime.h>
#include <math.h>

typedef __attribute__((ext_vector_type(2))) float v2f;
typedef __attribute__((ext_vector_type(8))) float v8f;

#define BATCH 512
#define IN    256
#define OUTN  1024

// s[b] = sum_i x[b,i]^2   (512 values)
__global__ __launch_bounds__(256)
void rowsq_kernel(const float* __restrict__ x, float* __restrict__ s) {
    int b = blockIdx.x * blockDim.x + threadIdx.x;
    if (b >= BATCH) return;
    const float* row = x + b * IN;
    float acc = 0.f;
#pragma unroll 8
    for (int i = 0; i < IN; ++i) { float v = row[i]; acc += v * v; }
    s[b] = acc;
}

// t[o] = sum_i w[i,o]^2   (1024 values, coalesced over o)
__global__ __launch_bounds__(256)
void colsq_kernel(const float* __restrict__ w, float* __restrict__ t) {
    int o = blockIdx.x * blockDim.x + threadIdx.x;
    if (o >= OUTN) return;
    float acc = 0.f;
#pragma unroll 8
    for (int i = 0; i < IN; ++i) { float v = w[i * OUTN + o]; acc += v * v; }
    t[o] = acc;
}

// out[b,o] = sqrt(s[b] + t[o] - 2*dot(x[b,:], w[:,o]))
// Dot via V_WMMA_F32_16X16X4_F32: one wave owns one 16x16 output tile.
// Block = 256 threads = 8 waves -> 16(M) x 128(N) macro-tile.
__global__ __launch_bounds__(256)
void euclid_wmma_kernel(const float* __restrict__ x,
                        const float* __restrict__ w,
                        const float* __restrict__ s,
                        const float* __restrict__ t,
                        float* __restrict__ out) {
    const int lane  = threadIdx.x & 31;
    const int wave  = threadIdx.x >> 5;
    const int tileM = blockIdx.y;                  // 0..31
    const int tileN = blockIdx.x * 8 + wave;       // 0..63
    const int m0 = tileM * 16;
    const int n0 = tileN * 16;

    const int idx15 = lane & 15;            // M-row for A frag, N-col for B/C frags
    const int khalf = (lane >> 4) * 2;      // K offset 0 or 2 per lane half

    // A: 16x4 tile of x (row-major, stride IN).  lane -> row m0+idx15, K base khalf.
    const float* aPtr = x + (size_t)(m0 + idx15) * IN + khalf;
    // B: 4x16 tile of w (row-major, stride OUTN). lane -> col n0+idx15, K base khalf.
    const float* bPtr = w + (size_t)khalf * OUTN + (n0 + idx15);

    v8f c = {};
#pragma unroll 8
    for (int k = 0; k < IN; k += 4) {
        v2f a, b;
        a.x = aPtr[0];            // K = khalf
        a.y = aPtr[1];            // K = khalf + 1
        b.x = bPtr[0];            // K = khalf
        b.y = bPtr[OUTN];         // K = khalf + 1
        aPtr += 4;
        bPtr += 4 * OUTN;
        // 8 args: (neg_a, A, neg_b, B, c_mod, C, reuse_a, reuse_b)
        c = __builtin_amdgcn_wmma_f32_16x16x4_f32(false, a, false, b,
                                                  (short)0, c, false, false);
    }

    // Epilogue: C/D layout -> lane holds N = n0+idx15, VGPR r holds M = r + 8*(lane>=16)
    const int   n  = n0 + idx15;
    const float tn = t[n];
    const int   mbase = m0 + ((lane >> 4) * 8);
#pragma unroll
    for (int r = 0; r < 8; ++r) {
        int m = mbase + r;
        float d2 = s[m] + tn - 2.0f * c[r];
        out[(size_t)m * OUTN + n] = sqrtf(fmaxf(d2, 0.0f));
    }
}

extern "C" void kernel_launch(void* const* d_in, const int* in_sizes, int n_in,
                              void* d_out, int out_size, void* d_ws, size_t ws_size,
                              hipStream_t stream) {
    const float* x = (const float*)d_in[0];   // [512, 256]
    const float* w = (const float*)d_in[1];   // [256, 1024]
    float* out = (float*)d_out;               // [512, 1024]

    float* s = (float*)d_ws;                  // 512 floats
    float* t = s + BATCH;                     // 1024 floats

    rowsq_kernel<<<BATCH / 256, 256, 0, stream>>>(x, s);
    colsq_kernel<<<OUTN / 256, 256, 0, stream>>>(w, t);
    euclid_wmma_kernel<<<dim3(OUTN / 16 / 8, BATCH / 16), 256, 0, stream>>>(x, w, s, t, out);
}